// FwdBwdFlowConsistency_81346680586607
// MI455X (gfx1250) — compile-verified
//
#include <hip/hip_runtime.h>
#include <hip/hip_bf16.h>

// Problem constants (fixed by the reference):
//   B=16, H=512, W=1024, C=2, flow layout [B,H,W,2] fp32.
//   loss = 0.5 * (num_f/den_f + num_b/den_b), each direction independent.
#define HW_SHIFT 19          // H*W = 512*1024 = 2^19 pixels per batch image
#define W_DIM    1024
#define H_DIM    512

__device__ __forceinline__ void sample_accum(const float2* __restrict__ other,
                                             int baseIdx, float2 f,
                                             float& num, float& den) {
    const float Wm1 = (float)(W_DIM - 1);
    const float Hm1 = (float)(H_DIM - 1);
    // grid -> pixel coords, border clamp, align_corners=True
    float x = fminf(fmaxf((f.x + 1.0f) * 0.5f * Wm1, 0.0f), Wm1);
    float y = fminf(fmaxf((f.y + 1.0f) * 0.5f * Hm1, 0.0f), Hm1);
    float x0f = floorf(x), y0f = floorf(y);
    float wx = x - x0f,  wy = y - y0f;
    int x0 = (int)x0f,   y0 = (int)y0f;
    int x1 = min(x0 + 1, W_DIM - 1);
    int y1 = min(y0 + 1, H_DIM - 1);
    int r0 = baseIdx + y0 * W_DIM;
    int r1 = baseIdx + y1 * W_DIM;
    // 4 bilinear corners, each a contiguous (x,y) pair -> global_load_b64
    float2 v00 = other[r0 + x0];
    float2 v01 = other[r0 + x1];
    float2 v10 = other[r1 + x0];
    float2 v11 = other[r1 + x1];
    float wx1 = 1.0f - wx, wy1 = 1.0f - wy;
    float sx = (v00.x * wx1 + v01.x * wx) * wy1 + (v10.x * wx1 + v11.x * wx) * wy;
    float sy = (v00.y * wx1 + v01.y * wx) * wy1 + (v10.y * wx1 + v11.y * wx) * wy;
    float dx = fabsf(sx + f.x);
    float dy = fabsf(sy + f.y);
    // mask: cnorm(diff) < max(0.05*cnorm(flow), 3)  <=>  squared form (no sqrt)
    float n2 = fmaf(dx, dx, dy * dy);
    float b2 = fmaxf(0.0025f * fmaf(f.x, f.x, f.y * f.y), 9.0f);
    float m  = (n2 < b2) ? 1.0f : 0.0f;
    num = fmaf(0.5f * (dx + dy), m, num);
    den += m;
}

__device__ __forceinline__ void block_reduce4_store(float v0, float v1, float v2, float v3,
                                                    float* __restrict__ dst4) {
    // wave32 butterfly reduce
    #pragma unroll
    for (int off = 16; off > 0; off >>= 1) {
        v0 += __shfl_xor(v0, off, 32);
        v1 += __shfl_xor(v1, off, 32);
        v2 += __shfl_xor(v2, off, 32);
        v3 += __shfl_xor(v3, off, 32);
    }
    __shared__ float red[8][4];
    int lane = threadIdx.x & 31;
    int wv   = threadIdx.x >> 5;
    if (lane == 0) {
        red[wv][0] = v0; red[wv][1] = v1; red[wv][2] = v2; red[wv][3] = v3;
    }
    __syncthreads();
    if (threadIdx.x == 0) {
        float s0 = 0.f, s1 = 0.f, s2 = 0.f, s3 = 0.f;
        #pragma unroll
        for (int w = 0; w < 8; ++w) {
            s0 += red[w][0]; s1 += red[w][1]; s2 += red[w][2]; s3 += red[w][3];
        }
        dst4[0] = s0; dst4[1] = s1; dst4[2] = s2; dst4[3] = s3;
    }
}

__global__ __launch_bounds__(256)
void flow_consistency_kernel(const float2* __restrict__ ff,
                             const float2* __restrict__ fb,
                             float* __restrict__ partials, int N) {
    float nf = 0.f, df = 0.f, nb = 0.f, db = 0.f;
    int stride = gridDim.x * blockDim.x;
    for (int i = blockIdx.x * blockDim.x + threadIdx.x; i < N; i += stride) {
        int ip = i + stride;
        if (ip < N) {                      // gfx1250 global_prefetch_b8 for streaming path
            __builtin_prefetch(&ff[ip], 0, 3);
            __builtin_prefetch(&fb[ip], 0, 3);
        }
        int base = (i >> HW_SHIFT) << HW_SHIFT;   // batch image base (pixel units)
        float2 vf = ff[i];
        float2 vb = fb[i];
        sample_accum(fb, base, vf, nf, df);       // forward direction
        sample_accum(ff, base, vb, nb, db);       // backward direction
    }
    block_reduce4_store(nf, df, nb, db, partials + (size_t)blockIdx.x * 4);
}

__global__ __launch_bounds__(256)
void flow_consistency_finalize(const float* __restrict__ partials, int nblocks,
                               float* __restrict__ out) {
    float v0 = 0.f, v1 = 0.f, v2 = 0.f, v3 = 0.f;
    for (int i = threadIdx.x; i < nblocks; i += 256) {
        const float* p = partials + (size_t)i * 4;
        v0 += p[0]; v1 += p[1]; v2 += p[2]; v3 += p[3];
    }
    #pragma unroll
    for (int off = 16; off > 0; off >>= 1) {
        v0 += __shfl_xor(v0, off, 32);
        v1 += __shfl_xor(v1, off, 32);
        v2 += __shfl_xor(v2, off, 32);
        v3 += __shfl_xor(v3, off, 32);
    }
    __shared__ float red[8][4];
    int lane = threadIdx.x & 31;
    int wv   = threadIdx.x >> 5;
    if (lane == 0) {
        red[wv][0] = v0; red[wv][1] = v1; red[wv][2] = v2; red[wv][3] = v3;
    }
    __syncthreads();
    if (threadIdx.x == 0) {
        float s0 = 0.f, s1 = 0.f, s2 = 0.f, s3 = 0.f;
        #pragma unroll
        for (int w = 0; w < 8; ++w) {
            s0 += red[w][0]; s1 += red[w][1]; s2 += red[w][2]; s3 += red[w][3];
        }
        out[0] = 0.5f * (s0 / s1 + s2 / s3);
    }
}

extern "C" void kernel_launch(void* const* d_in, const int* in_sizes, int n_in,
                              void* d_out, int out_size, void* d_ws, size_t ws_size,
                              hipStream_t stream) {
    const float2* ff = (const float2*)d_in[0];   // flow_fwd [B,H,W,2]
    const float2* fb = (const float2*)d_in[1];   // flow_bwd [B,H,W,2]
    float* partials  = (float*)d_ws;             // BLOCKS * 4 floats
    float* out       = (float*)d_out;            // scalar fp32

    int N = in_sizes[0] / 2;                     // pixel count = B*H*W
    const int THREADS = 256;
    const int BLOCKS  = 1024;                    // 262144 threads, 32 px/thread/direction

    flow_consistency_kernel<<<BLOCKS, THREADS, 0, stream>>>(ff, fb, partials, N);
    flow_consistency_finalize<<<1, THREADS, 0, stream>>>(partials, BLOCKS, out);
}